// HWLRMF_64441689309486
// MI455X (gfx1250) — compile-verified
//
#include <hip/hip_runtime.h>
#include <math.h>

typedef __attribute__((ext_vector_type(2))) float v2f;
typedef __attribute__((ext_vector_type(8))) float v8f;

constexpr int BAND = 64;
constexpr int CH   = 64;
constexpr int NPIX = 16384;               // 128*128
constexpr int ITE  = 20;
constexpr long long TOT = (long long)BAND * CH * NPIX;   // 67,108,864

// ---------------------------------------------------------------- reductions
__device__ inline void blockReduceAtomic(float v, float* target) {
    for (int off = 16; off > 0; off >>= 1)
        v += __shfl_down(v, off, 32);                    // wave32 reduce
    __shared__ float wsum[8];
    const int lane = threadIdx.x & 31, wid = threadIdx.x >> 5;
    if (lane == 0) wsum[wid] = v;
    __syncthreads();
    if (threadIdx.x == 0) {
        float s = 0.f;
        const int nw = (blockDim.x + 31) >> 5;
        for (int i = 0; i < nw; ++i) s += wsum[i];
        atomicAdd(target, s);
    }
    __syncthreads();
}

// ---------------------------------------------------------------- small init
__global__ void init_misc(float* sumW, float* lossF, float* loss) {
    const int t = threadIdx.x;
    if (t == 0) sumW[0] = 0.f;
    if (t < ITE) { lossF[t] = 0.f; loss[t] = 0.f; }
}

__global__ void zero_buf(float* p, long long n) {
    const long long i = (long long)blockIdx.x * blockDim.x + threadIdx.x;
    if (i < n) p[i] = 0.f;
}

__global__ __launch_bounds__(256)
void sum_reduce(const float* __restrict__ W, float* __restrict__ out) {
    long long i = (long long)blockIdx.x * blockDim.x + threadIdx.x;
    const long long stride = (long long)gridDim.x * blockDim.x;
    float v = 0.f;
    for (; i < TOT; i += stride) v += W[i];
    blockReduceAtomic(v, out);
}

// ---------------------------------------------------------------- G = L L^T
// Per band: 64x64 output, K = 16384, split 8 ways across workgroups.
// 512 threads = 16 waves; wave w owns 16x16 tile (w/4, w%4).
// V_WMMA_F32_16X16X4_F32 per ISA layout:
//   A 16x4: lanes 0-15 -> M=lane, v0=K0,v1=K1 ; lanes 16-31 -> v0=K2,v1=K3
//   B 4x16: lanes 0-15 -> N=lane, K lo pair  ; lanes 16-31 -> K hi pair
//   C/D   : vgpr r, lanes 0-15 -> M=r, N=lane; lanes 16-31 -> M=8+r
__global__ __launch_bounds__(512)
void gemm_G(const float* __restrict__ Lsrc, float* __restrict__ G) {
    constexpr int KSPLIT = 8;
    constexpr int KCHUNK = NPIX / KSPLIT;          // 2048
    const int band   = blockIdx.x / KSPLIT;
    const int kchunk = blockIdx.x % KSPLIT;
    const long long base = (long long)band * CH * NPIX + (long long)kchunk * KCHUNK;

    __shared__ float lds[64][68];                   // padded stride

    const int tid  = threadIdx.x;
    const int wave = tid >> 5;
    const int lane = tid & 31;
    const int ti   = wave >> 2, tj = wave & 3;
    const int m    = lane & 15;
    const int half = lane >> 4;

    const int ldch = tid >> 3;                      // 0..63 row to load
    const int ldk0 = (tid & 7) * 8;                 // 8 floats per thread

    v8f acc = {};

    for (int kb = 0; kb < KCHUNK; kb += 64) {
        const float* src = Lsrc + base + (long long)ldch * NPIX + kb + ldk0;
        float4 p0 = *(const float4*)(src);
        float4 p1 = *(const float4*)(src + 4);
        __builtin_prefetch(src + 64, 0, 1);         // global_prefetch_b8 (next block)
        __syncthreads();
        *(float4*)&lds[ldch][ldk0]     = p0;
        *(float4*)&lds[ldch][ldk0 + 4] = p1;
        __syncthreads();

        #pragma unroll
        for (int kk = 0; kk < 64; kk += 4) {
            const int ko = kk + 2 * half;
            v2f a, b;
            a.x = lds[16 * ti + m][ko];
            a.y = lds[16 * ti + m][ko + 1];
            b.x = lds[16 * tj + m][ko];
            b.y = lds[16 * tj + m][ko + 1];
            acc = __builtin_amdgcn_wmma_f32_16x16x4_f32(
                false, a, false, b, (short)0, acc, false, false);
        }
    }

    const int col = 16 * tj + m;
    #pragma unroll
    for (int r = 0; r < 8; ++r) {
        const int row = 16 * ti + r + 8 * half;
        atomicAdd(&G[band * 4096 + row * 64 + col], acc[r]);
    }
}

// ---------------------------------------------------------------- Jacobi eig
// One workgroup (64 threads) per band. Parallel cyclic Jacobi on 64x64
// symmetric PSD G in LDS; round-robin tournament gives 32 disjoint pairs per
// round. Extract top-3 eigenvectors -> Ur[band][c][r].
__global__ __launch_bounds__(64)
void jacobi_topr(const float* __restrict__ G, float* __restrict__ Ur) {
    __shared__ float A[64][65];
    __shared__ float V[64][65];
    __shared__ int   top[32], bot[32];
    __shared__ float cs[32], sn[32];
    __shared__ int   idx3[3];

    const int band = blockIdx.x;
    const int tid  = threadIdx.x;

    for (int r = 0; r < 64; ++r) {
        A[r][tid] = G[band * 4096 + r * 64 + tid];
        V[r][tid] = (r == tid) ? 1.f : 0.f;
    }
    if (tid < 32) { top[tid] = tid; bot[tid] = 32 + tid; }

    constexpr int NSWEEP = 10;
    for (int sweep = 0; sweep < NSWEEP; ++sweep) {
        for (int round = 0; round < 63; ++round) {
            __syncthreads();
            if (tid < 32) {
                const int p = top[tid], q = bot[tid];
                const float app = A[p][p], aqq = A[q][q], apq = A[p][q];
                float c = 1.f, s = 0.f;
                if (fabsf(apq) > 1e-12f) {
                    const float tau = (aqq - app) / (2.f * apq);
                    const float t = (tau >= 0.f ? 1.f : -1.f) /
                                    (fabsf(tau) + sqrtf(1.f + tau * tau));
                    c = rsqrtf(1.f + t * t);
                    s = t * c;
                }
                cs[tid] = c; sn[tid] = s;
            }
            __syncthreads();
            // column update (A <- A J, V <- V J); thread owns row `tid`
            for (int t = 0; t < 32; ++t) {
                const int p = top[t], q = bot[t];
                const float c = cs[t], s = sn[t];
                const float gp = A[tid][p], gq = A[tid][q];
                A[tid][p] = c * gp - s * gq;
                A[tid][q] = s * gp + c * gq;
                const float vp = V[tid][p], vq = V[tid][q];
                V[tid][p] = c * vp - s * vq;
                V[tid][q] = s * vp + c * vq;
            }
            __syncthreads();
            // row update (A <- J^T A); thread owns column `tid`
            for (int t = 0; t < 32; ++t) {
                const int p = top[t], q = bot[t];
                const float c = cs[t], s = sn[t];
                const float gp = A[p][tid], gq = A[q][tid];
                A[p][tid] = c * gp - s * gq;
                A[q][tid] = s * gp + c * gq;
            }
            __syncthreads();
            if (tid == 0) {                          // rotate tournament
                const int lastTop = top[31], firstBot = bot[0];
                for (int i = 31; i >= 2; --i) top[i] = top[i - 1];
                for (int i = 0; i < 31; ++i)  bot[i] = bot[i + 1];
                top[1]  = firstBot;
                bot[31] = lastTop;
            }
        }
    }
    __syncthreads();

    if (tid == 0) {                                  // top-3 eigenvalues
        float best[3] = {-1e30f, -1e30f, -1e30f};
        int   bi[3]   = {0, 0, 0};
        for (int i = 0; i < 64; ++i) {
            const float d = A[i][i];
            if (d > best[0]) { best[2]=best[1]; bi[2]=bi[1]; best[1]=best[0]; bi[1]=bi[0]; best[0]=d; bi[0]=i; }
            else if (d > best[1]) { best[2]=best[1]; bi[2]=bi[1]; best[1]=d; bi[1]=i; }
            else if (d > best[2]) { best[2]=d; bi[2]=i; }
        }
        idx3[0]=bi[0]; idx3[1]=bi[1]; idx3[2]=bi[2];
    }
    __syncthreads();
    #pragma unroll
    for (int r = 0; r < 3; ++r)
        Ur[(band * 64 + tid) * 3 + r] = V[tid][idx3[r]];
}

// ---------------------------------------------------------------- UtL = Ur^T L
__global__ __launch_bounds__(256)
void gemm_UtL(const float* __restrict__ Lsrc, const float* __restrict__ Ur,
              float* __restrict__ UtL) {
    const int band = blockIdx.x >> 6;                // 64 n-blocks per band
    const int nblk = blockIdx.x & 63;
    const int n    = nblk * 256 + threadIdx.x;

    __shared__ float u[64][3];
    if (threadIdx.x < 192)
        ((float*)u)[threadIdx.x] = Ur[band * 192 + threadIdx.x];
    __syncthreads();

    const float* Lp = Lsrc + (long long)band * CH * NPIX + n;
    float a0 = 0.f, a1 = 0.f, a2 = 0.f;
    #pragma unroll 4
    for (int c = 0; c < 64; ++c) {
        const float v = Lp[(long long)c * NPIX];     // coalesced across lanes
        a0 = fmaf(u[c][0], v, a0);
        a1 = fmaf(u[c][1], v, a1);
        a2 = fmaf(u[c][2], v, a2);
    }
    float* Up = UtL + (long long)band * 3 * NPIX + n;
    Up[0]        = a0;
    Up[NPIX]     = a1;
    Up[2 * NPIX] = a2;
}

// ---------------------------------------------------------------- fused tail
// UV = Ur*UtL (rank-3, no 268MB UV buffer), losses, mu-blend, L update.
__global__ __launch_bounds__(256)
void fused_update(const float* __restrict__ x, const float* __restrict__ W,
                  const float* __restrict__ Ur, const float* __restrict__ UtL,
                  const float* __restrict__ sumW, float* __restrict__ Lout,
                  float* __restrict__ out_uv, float* __restrict__ out_lossF,
                  float* __restrict__ out_loss, float alpha_pow, int write_uv) {
    const long long e = (long long)blockIdx.x * 256 + threadIdx.x;
    const int band = (int)(e >> 20);                 // CH*NPIX = 1<<20
    const int c    = (int)((e >> 14) & 63);          // NPIX = 1<<14
    const int n    = (int)(e & (NPIX - 1));

    const float u0 = Ur[(band * 64 + c) * 3 + 0];    // uniform per block
    const float u1 = Ur[(band * 64 + c) * 3 + 1];
    const float u2 = Ur[(band * 64 + c) * 3 + 2];
    const float* Up = UtL + (long long)band * 3 * NPIX + n;
    const float uv = u0 * Up[0] + u1 * Up[NPIX] + u2 * Up[2 * NPIX];

    const float xv = x[e], wv = W[e];
    const float rho = 0.5f * (sumW[0] / (float)TOT) * alpha_pow;
    const float mu  = rho / (wv + rho);
    Lout[e] = (1.f - mu) * xv + mu * uv;
    if (write_uv) out_uv[e] = uv;

    const float d  = uv - xv;
    const float inv = 1.f / (float)TOT;
    blockReduceAtomic(uv * uv * inv, out_loss);
    blockReduceAtomic(wv * d * d * inv, out_lossF);
}

// ---------------------------------------------------------------- host driver
extern "C" void kernel_launch(void* const* d_in, const int* in_sizes, int n_in,
                              void* d_out, int out_size, void* d_ws, size_t ws_size,
                              hipStream_t stream) {
    const float* x = (const float*)d_in[0];
    const float* W = (const float*)d_in[1];

    float* out      = (float*)d_out;
    float* out_uv   = out;                 // (64,64,128,128)
    float* out_lossF= out + TOT;           // (20,)
    float* out_loss = out + TOT + ITE;     // (20,)

    float* ws  = (float*)d_ws;
    float* L   = ws;                                           // 67,108,864
    float* G   = L   + TOT;                                    // 262,144
    float* Ur  = G   + 64 * 64 * 64;                           // 12,288
    float* UtL = Ur  + 64 * 64 * 3;                            // 3,145,728
    float* sW  = UtL + (long long)64 * 3 * NPIX;               // 1

    init_misc<<<1, 64, 0, stream>>>(sW, out_lossF, out_loss);
    sum_reduce<<<2048, 256, 0, stream>>>(W, sW);

    double ap = 1.0;
    for (int k = 0; k < ITE; ++k) {
        const float* Lsrc = (k == 0) ? x : L;
        zero_buf<<<(64 * 64 * 64 + 255) / 256, 256, 0, stream>>>(G, 64LL * 64 * 64);
        gemm_G<<<64 * 8, 512, 0, stream>>>(Lsrc, G);
        jacobi_topr<<<64, 64, 0, stream>>>(G, Ur);
        gemm_UtL<<<64 * 64, 256, 0, stream>>>(Lsrc, Ur, UtL);
        fused_update<<<(int)(TOT / 256), 256, 0, stream>>>(
            x, W, Ur, UtL, sW, L, out_uv, out_lossF + k, out_loss + k,
            (float)ap, (k == ITE - 1) ? 1 : 0);
        ap *= 1.05;
    }
}